// GCN_LPA_88321707475505
// MI455X (gfx1250) — compile-verified
//
#include <hip/hip_runtime.h>
#include <hip/hip_bf16.h>

// ---------------------------------------------------------------------------
// GCN + LPA for MI455X (gfx1250, wave32).
//  - LPA / edge aggregation: float4 gathers + global_atomic_add_f32 (L2-resident)
//  - Dense projections: V_WMMA_F32_16X16X4_F32, one wave per 16x32 tile
//    (two accumulators share each A fragment -> 2 WMMAs per A load)
//  - log_softmax: one wave32 per node, __shfl_xor butterfly reductions
// ---------------------------------------------------------------------------

typedef __attribute__((ext_vector_type(2))) float v2f;
typedef __attribute__((ext_vector_type(8))) float v8f;

#define IN_DIM 256
#define HID    128
#define NCLASS 64
#define LPA_ITERS 20

// -------------------------------------------------- utility kernels
__global__ void zero_kernel(float* __restrict__ p, long long n) {
  long long i = (long long)blockIdx.x * blockDim.x + threadIdx.x;
  if (i < n) p[i] = 0.0f;
}

__global__ void degrees_kernel(const int* __restrict__ src, const int* __restrict__ dst,
                               float* __restrict__ out_deg, float* __restrict__ in_deg, int E) {
  int i = blockIdx.x * blockDim.x + threadIdx.x;
  if (i >= E) return;
  unsafeAtomicAdd(&out_deg[src[i]], 1.0f);
  unsafeAtomicAdd(&in_deg[dst[i]], 1.0f);
}

__global__ void norms_kernel(const float* __restrict__ in_deg, const float* __restrict__ out_deg,
                             float* __restrict__ inv_in, float* __restrict__ norm_src,
                             float* __restrict__ norm_dst, int N) {
  int i = blockIdx.x * blockDim.x + threadIdx.x;
  if (i >= N) return;
  float ind = fmaxf(in_deg[i], 1.0f);
  float outd = fmaxf(out_deg[i], 1.0f);
  inv_in[i] = 1.0f / ind;
  norm_dst[i] = rsqrtf(ind);
  norm_src[i] = rsqrtf(outd);
}

// -------------------------------------------------- LPA
__global__ void lpa_init_kernel(float* __restrict__ emb, const float* __restrict__ labels,
                                const unsigned char* __restrict__ train, int N) {
  long long t = (long long)blockIdx.x * blockDim.x + threadIdx.x;
  if (t >= (long long)N * NCLASS) return;
  int n = (int)(t >> 6);
  emb[t] = train[n] ? labels[t] : 0.0f;
}

__global__ void lpa_update_kernel(float* __restrict__ agg, const float* __restrict__ labels,
                                  const unsigned char* __restrict__ train,
                                  const float* __restrict__ inv_in, int N) {
  long long t = (long long)blockIdx.x * blockDim.x + threadIdx.x;
  if (t >= (long long)N * NCLASS) return;
  int n = (int)(t >> 6);
  agg[t] = train[n] ? labels[t] : agg[t] * inv_in[n];  // in-place: agg -> new emb
}

// -------------------------------------------------- edge scatter (DIM = 64 or 128)
// TPE = DIM/32 threads per edge; each thread moves 32 floats via 8 float4 loads.
template <int DIM>
__global__ void edge_scatter_kernel(const float* __restrict__ x, const int* __restrict__ src,
                                    const int* __restrict__ dst, const float* __restrict__ ew,
                                    float* __restrict__ agg, int E) {
  const int TPE = DIM / 32;
  long long t = (long long)blockIdx.x * blockDim.x + threadIdx.x;
  long long e = t / TPE;
  if (e >= E) return;
  int part = (int)(t % TPE);
  int s = src[e];
  int d = dst[e];
  float w = ew[e];
  const float4* xs = (const float4*)(x + (size_t)s * DIM + part * 32);
  float* ad = agg + (size_t)d * DIM + part * 32;
#pragma unroll
  for (int i = 0; i < 8; ++i) {
    float4 v = xs[i];
    unsafeAtomicAdd(ad + i * 4 + 0, v.x * w);
    unsafeAtomicAdd(ad + i * 4 + 1, v.y * w);
    unsafeAtomicAdd(ad + i * 4 + 2, v.z * w);
    unsafeAtomicAdd(ad + i * 4 + 3, v.w * w);
  }
}

// -------------------------------------------------- WMMA fp32 matmul
// C[N x M] = (A[N x K] row-scaled by rs) @ W[K x M].  One wave per 16x32 tile:
// two 16x16 accumulators share each A fragment (2 WMMAs per A load).
// A fragment (16x4, 2 VGPRs):  m = lane&15, kb = (lane>>4)*2 -> {A[m][kb], A[m][kb+1]}
// B fragment (4x16, 2 VGPRs):  n = lane&15, kb              -> {W[kb][n], W[kb+1][n]}
// C/D (16x16, 8 VGPRs):        n = lane&15, rows (lane>>4)*8 + v
template <int K>
__global__ void wmma_rowscale_matmul(const float* __restrict__ A, const float* __restrict__ rs,
                                     const float* __restrict__ W, float* __restrict__ C,
                                     int Nrows, int Mcols) {
  int wave = blockIdx.x * (blockDim.x >> 5) + (threadIdx.x >> 5);
  int lane = threadIdx.x & 31;
  int colPairs = Mcols >> 5;  // 16x32 tiles per row band
  int rowTile = wave / colPairs;
  int colPair = wave % colPairs;
  int row0 = rowTile * 16;
  if (row0 >= Nrows) return;  // wave-uniform: EXEC stays all-ones for WMMA

  int mn = lane & 15;            // A row within tile / B-C column within sub-tile
  int kb = (lane >> 4) << 1;     // 0 or 2
  int row = row0 + mn;
  row = row < Nrows ? row : Nrows - 1;  // defensive clamp (tiles are full in practice)
  float scale = rs[row];
  const float* arow = A + (size_t)row * K;
  int col0 = colPair * 32;
  const float* wcol0 = W + col0 + mn;        // left 16 columns
  const float* wcol1 = W + col0 + 16 + mn;   // right 16 columns

  v8f acc0 = {0.f, 0.f, 0.f, 0.f, 0.f, 0.f, 0.f, 0.f};
  v8f acc1 = {0.f, 0.f, 0.f, 0.f, 0.f, 0.f, 0.f, 0.f};
#pragma unroll 4
  for (int k = 0; k < K; k += 4) {
    __builtin_prefetch(arow + k + 64, 0, 1);  // global_prefetch_b8
    v2f a;
    a.x = arow[k + kb] * scale;
    a.y = arow[k + kb + 1] * scale;
    size_t r0 = (size_t)(k + kb) * Mcols;
    size_t r1 = (size_t)(k + kb + 1) * Mcols;
    v2f b0; b0.x = wcol0[r0]; b0.y = wcol0[r1];
    v2f b1; b1.x = wcol1[r0]; b1.y = wcol1[r1];
    acc0 = __builtin_amdgcn_wmma_f32_16x16x4_f32(false, a, false, b0, (short)0, acc0, false, false);
    acc1 = __builtin_amdgcn_wmma_f32_16x16x4_f32(false, a, false, b1, (short)0, acc1, false, false);
  }

  int mbase = (lane >> 4) * 8;
  float* cbase = C + col0 + mn;
  if (row0 + 16 <= Nrows) {  // full tile: unconditional, branch-free stores
    float* cp = cbase + (size_t)(row0 + mbase) * Mcols;
#pragma unroll
    for (int v = 0; v < 8; ++v) {
      cp[0] = acc0[v];
      cp[16] = acc1[v];
      cp += Mcols;
    }
  } else {  // ragged tail (not hit for N % 16 == 0, kept for generality)
#pragma unroll
    for (int v = 0; v < 8; ++v) {
      int r = row0 + mbase + v;
      if (r < Nrows) {
        cbase[(size_t)r * Mcols] = acc0[v];
        cbase[(size_t)r * Mcols + 16] = acc1[v];
      }
    }
  }
}

// -------------------------------------------------- finish layer-1: relu(agg*norm_dst + b1)
__global__ void finish1_kernel(const float* __restrict__ agg, const float* __restrict__ norm_dst,
                               const float* __restrict__ b1, float* __restrict__ x1, int N) {
  long long t = (long long)blockIdx.x * blockDim.x + threadIdx.x;
  if (t >= (long long)N * HID) return;
  int n = (int)(t >> 7);
  int d = (int)(t & (HID - 1));
  float v = agg[t] * norm_dst[n] + b1[d];
  x1[t] = v > 0.0f ? v : 0.0f;
}

// -------------------------------------------------- log_softmax over 64 classes, one wave/node
// optional per-row scale and per-class bias (pass nullptr for identity)
__global__ void logsoftmax64_kernel(const float* __restrict__ x, const float* __restrict__ scale,
                                    const float* __restrict__ bias, float* __restrict__ out, int N) {
  int n = blockIdx.x * (blockDim.x >> 5) + (threadIdx.x >> 5);
  if (n >= N) return;  // wave-uniform
  int lane = threadIdx.x & 31;
  float s = scale ? scale[n] : 1.0f;
  float bA = bias ? bias[lane] : 0.0f;
  float bB = bias ? bias[32 + lane] : 0.0f;
  const float* xr = x + (size_t)n * NCLASS;
  float v0 = xr[lane] * s + bA;
  float v1 = xr[32 + lane] * s + bB;
  float m = fmaxf(v0, v1);
#pragma unroll
  for (int o = 16; o > 0; o >>= 1) m = fmaxf(m, __shfl_xor(m, o, 32));
  float sum = __expf(v0 - m) + __expf(v1 - m);
#pragma unroll
  for (int o = 16; o > 0; o >>= 1) sum += __shfl_xor(sum, o, 32);
  float lse = m + __logf(sum);
  float* outr = out + (size_t)n * NCLASS;
  outr[lane] = v0 - lse;
  outr[32 + lane] = v1 - lse;
}

// -------------------------------------------------- host orchestration
static inline int cdiv(long long a, int b) { return (int)((a + b - 1) / b); }

extern "C" void kernel_launch(void* const* d_in, const int* in_sizes, int n_in,
                              void* d_out, int out_size, void* d_ws, size_t ws_size,
                              hipStream_t stream) {
  const float* h = (const float*)d_in[0];
  const float* labels = (const float*)d_in[1];
  const int* src = (const int*)d_in[2];
  const int* dst = (const int*)d_in[3];
  const unsigned char* train = (const unsigned char*)d_in[4];
  const float* ew = (const float*)d_in[5];
  const float* W1 = (const float*)d_in[6];
  const float* b1 = (const float*)d_in[7];
  const float* W2 = (const float*)d_in[8];
  const float* b2 = (const float*)d_in[9];

  const int N = in_sizes[4];  // train is (N,)
  const int E = in_sizes[2];  // src is (E,)
  float* outp = (float*)d_out;

  // workspace layout (floats)
  float* ws = (float*)d_ws;
  float* in_deg = ws;                 // N
  float* out_deg = ws + (size_t)N;    // N
  float* inv_in = ws + 2ll * N;       // N
  float* norm_src = ws + 3ll * N;     // N
  float* norm_dst = ws + 4ll * N;     // N
  float* bufA = ws + 5ll * N;                    // N*128
  float* bufB = bufA + (size_t)N * HID;          // N*128

  const int B = 256;

  // degrees + norms
  zero_kernel<<<cdiv(2ll * N, B), B, 0, stream>>>(in_deg, 2ll * N);
  degrees_kernel<<<cdiv(E, B), B, 0, stream>>>(src, dst, out_deg, in_deg, E);
  norms_kernel<<<cdiv(N, B), B, 0, stream>>>(in_deg, out_deg, inv_in, norm_src, norm_dst, N);

  // ---------------- LPA (ping-pong bufA <-> bufB, first N*64 floats of each)
  float* emb = bufA;
  float* agg = bufB;
  lpa_init_kernel<<<cdiv((long long)N * NCLASS, B), B, 0, stream>>>(emb, labels, train, N);
  for (int it = 0; it < LPA_ITERS; ++it) {
    zero_kernel<<<cdiv((long long)N * NCLASS, B), B, 0, stream>>>(agg, (long long)N * NCLASS);
    edge_scatter_kernel<NCLASS><<<cdiv((long long)E * (NCLASS / 32), B), B, 0, stream>>>(
        emb, src, dst, ew, agg, E);
    lpa_update_kernel<<<cdiv((long long)N * NCLASS, B), B, 0, stream>>>(agg, labels, train, inv_in, N);
    float* tmp = emb; emb = agg; agg = tmp;  // agg (updated in-place) becomes new emb
  }
  // second output: log_softmax(emb)
  logsoftmax64_kernel<<<cdiv(N, B / 32), B, 0, stream>>>(emb, nullptr, nullptr,
                                                         outp + (size_t)N * NCLASS, N);
  float* z1 = agg;     // free N*128 buffer (old ping-pong partner)
  float* agg1 = emb;   // emb consumed -> reuse as aggregation target

  // ---------------- GCN layer 1: z1 = (h * norm_src) @ W1 ; agg over edges ; relu(+b1)
  {
    long long waves = (long long)cdiv(N, 16) * (HID / 32);
    wmma_rowscale_matmul<IN_DIM><<<cdiv(waves * 32, 128), 128, 0, stream>>>(h, norm_src, W1, z1, N, HID);
  }
  zero_kernel<<<cdiv((long long)N * HID, B), B, 0, stream>>>(agg1, (long long)N * HID);
  edge_scatter_kernel<HID><<<cdiv((long long)E * (HID / 32), B), B, 0, stream>>>(
      z1, src, dst, ew, agg1, E);
  float* x1 = z1;  // overwrite z1 with relu output
  finish1_kernel<<<cdiv((long long)N * HID, B), B, 0, stream>>>(agg1, norm_dst, b1, x1, N);

  // ---------------- GCN layer 2: z2 = (x1 * norm_src) @ W2 ; agg ; log_softmax(+b2, *norm_dst)
  float* z2 = agg1;  // N*64 fits in freed buffer
  {
    long long waves = (long long)cdiv(N, 16) * (NCLASS / 32);
    wmma_rowscale_matmul<HID><<<cdiv(waves * 32, 128), 128, 0, stream>>>(x1, norm_src, W2, z2, N, NCLASS);
  }
  float* agg2 = x1;  // x1 consumed by matmul2 -> reuse
  zero_kernel<<<cdiv((long long)N * NCLASS, B), B, 0, stream>>>(agg2, (long long)N * NCLASS);
  edge_scatter_kernel<NCLASS><<<cdiv((long long)E * (NCLASS / 32), B), B, 0, stream>>>(
      z2, src, dst, ew, agg2, E);
  // first output: log_softmax(agg2 * norm_dst + b2)
  logsoftmax64_kernel<<<cdiv(N, B / 32), B, 0, stream>>>(agg2, norm_dst, b2, outp, N);
}